// FlexibleArgmax_23467701305396
// MI455X (gfx1250) — compile-verified
//
#include <hip/hip_runtime.h>
#include <stdint.h>

// FlexibleArgmax on MI455X (gfx1250, wave32).
// Pass 1: per-block 8192-entry u64 LDS scoreboard (64 KB of the 320 KB WGP LDS),
//         key = mono(x)<<32 | ~idx, ds_max_u64 per item, global_atomic_max_u64 flush.
// Pass 2: async-copy best[] (64 KB) global->LDS via GLOBAL_LOAD_ASYNC_TO_LDS_B128,
//         build winner-index thresholds, ds_add_u32 count of items before winner,
//         flush with global atomic add directly into d_out (== within-group rank).
// Traffic floor: 128 MB (pass1) + 64 MB (pass2, L2-resident) ~= 8.3 us @ 23.3 TB/s.

#define N_GROUPS 8192
#define TPB      256
#define NB_MAIN  256

// Monotone mapping: float total order -> unsigned total order.
__device__ __forceinline__ uint32_t mono_f32(float f) {
    uint32_t b = __float_as_uint(f);
    return (b & 0x80000000u) ? ~b : (b | 0x80000000u);
}

__global__ void fa_init(unsigned long long* __restrict__ best, int* __restrict__ out) {
    int g = blockIdx.x * blockDim.x + threadIdx.x;
    if (g < N_GROUPS) {
        best[g] = 0ull;   // below any real encoded key
        out[g]  = 0;
    }
}

__global__ __launch_bounds__(TPB)
void fa_pass1(const float* __restrict__ x,
              const int*   __restrict__ gi,
              unsigned long long* __restrict__ best,
              int nvec, int n) {
    __shared__ unsigned long long sBest[N_GROUPS];   // 64 KB
    for (int g = threadIdx.x; g < N_GROUPS; g += TPB) sBest[g] = 0ull;
    __syncthreads();

    const float4* __restrict__ x4 = (const float4*)x;
    const int4*   __restrict__ g4 = (const int4*)gi;
    const int stride = gridDim.x * TPB;

    int v = blockIdx.x * TPB + threadIdx.x;
    // 2x unrolled stream: 4 b128 loads in flight per wave iteration.
    for (; v + stride < nvec; v += 2 * stride) {
        float4 xa = x4[v];
        int4   ga = g4[v];
        float4 xb = x4[v + stride];
        int4   gb = g4[v + stride];
        uint32_t ba = (uint32_t)v * 4u;
        uint32_t bb = (uint32_t)(v + stride) * 4u;
        atomicMax(&sBest[ga.x], ((unsigned long long)mono_f32(xa.x) << 32) | (uint32_t)~(ba + 0u));
        atomicMax(&sBest[ga.y], ((unsigned long long)mono_f32(xa.y) << 32) | (uint32_t)~(ba + 1u));
        atomicMax(&sBest[ga.z], ((unsigned long long)mono_f32(xa.z) << 32) | (uint32_t)~(ba + 2u));
        atomicMax(&sBest[ga.w], ((unsigned long long)mono_f32(xa.w) << 32) | (uint32_t)~(ba + 3u));
        atomicMax(&sBest[gb.x], ((unsigned long long)mono_f32(xb.x) << 32) | (uint32_t)~(bb + 0u));
        atomicMax(&sBest[gb.y], ((unsigned long long)mono_f32(xb.y) << 32) | (uint32_t)~(bb + 1u));
        atomicMax(&sBest[gb.z], ((unsigned long long)mono_f32(xb.z) << 32) | (uint32_t)~(bb + 2u));
        atomicMax(&sBest[gb.w], ((unsigned long long)mono_f32(xb.w) << 32) | (uint32_t)~(bb + 3u));
    }
    for (; v < nvec; v += stride) {
        float4 xv = x4[v];
        int4   gv = g4[v];
        uint32_t base = (uint32_t)v * 4u;
        atomicMax(&sBest[gv.x], ((unsigned long long)mono_f32(xv.x) << 32) | (uint32_t)~(base + 0u));
        atomicMax(&sBest[gv.y], ((unsigned long long)mono_f32(xv.y) << 32) | (uint32_t)~(base + 1u));
        atomicMax(&sBest[gv.z], ((unsigned long long)mono_f32(xv.z) << 32) | (uint32_t)~(base + 2u));
        atomicMax(&sBest[gv.w], ((unsigned long long)mono_f32(xv.w) << 32) | (uint32_t)~(base + 3u));
    }
    // scalar tail (n not multiple of 4)
    for (int i = nvec * 4 + blockIdx.x * TPB + (int)threadIdx.x; i < n; i += stride) {
        atomicMax(&sBest[gi[i]],
                  ((unsigned long long)mono_f32(x[i]) << 32) | (uint32_t)~((uint32_t)i));
    }
    __syncthreads();

    for (int g = threadIdx.x; g < N_GROUPS; g += TPB) {
        unsigned long long k = sBest[g];
        if (k) atomicMax(&best[g], k);   // global_atomic_max_u64, 8192 lines in L2
    }
}

__global__ __launch_bounds__(TPB)
void fa_pass2(const int* __restrict__ gi,
              const unsigned long long* __restrict__ best,
              int* __restrict__ out,
              int nvec, int n) {
    __shared__ unsigned long long sBestRaw[N_GROUPS];  // 64 KB, async-copy target
    __shared__ uint32_t sThr[N_GROUPS];                // winner global index (32 KB)
    __shared__ uint32_t sCnt[N_GROUPS];                // items before winner (32 KB)

    // --- Async bulk copy best[] (64 KB) global -> LDS, ASYNCcnt-tracked. ---
    // Low 32 bits of a generic pointer to __shared__ == LDS byte offset
    // (flat-aperture mapping, ISA 10.2), which is what VDST of
    // global_load_async_to_lds_* expects.
    {
        uint32_t lds_base = (uint32_t)(uintptr_t)(void*)&sBestRaw[0];
        // 64 KB / 16 B = 4096 b128 transfers; 16 per thread.
        for (int k = 0; k < 16; ++k) {
            uint32_t byteOff = ((uint32_t)threadIdx.x + (uint32_t)k * TPB) * 16u;
            asm volatile("global_load_async_to_lds_b128 %0, %1, %2"
                         :
                         : "v"(lds_base + byteOff), "v"(byteOff), "s"(best)
                         : "memory");
        }
#if __has_builtin(__builtin_amdgcn_s_wait_asynccnt)
        __builtin_amdgcn_s_wait_asynccnt(0);
#else
        asm volatile("s_wait_asynccnt 0x0" ::: "memory");
#endif
    }
    __syncthreads();

    for (int g = threadIdx.x; g < N_GROUPS; g += TPB) {
        sThr[g] = ~(uint32_t)(sBestRaw[g] & 0xffffffffull);  // idx = ~low32(key)
        sCnt[g] = 0u;
    }
    __syncthreads();

    const int4* __restrict__ g4 = (const int4*)gi;
    const int stride = gridDim.x * TPB;

    int v = blockIdx.x * TPB + threadIdx.x;
    for (; v + stride < nvec; v += 2 * stride) {
        int4 ga = g4[v];
        int4 gb = g4[v + stride];
        uint32_t ba = (uint32_t)v * 4u;
        uint32_t bb = (uint32_t)(v + stride) * 4u;
        if (ba + 0u < sThr[ga.x]) atomicAdd(&sCnt[ga.x], 1u);
        if (ba + 1u < sThr[ga.y]) atomicAdd(&sCnt[ga.y], 1u);
        if (ba + 2u < sThr[ga.z]) atomicAdd(&sCnt[ga.z], 1u);
        if (ba + 3u < sThr[ga.w]) atomicAdd(&sCnt[ga.w], 1u);
        if (bb + 0u < sThr[gb.x]) atomicAdd(&sCnt[gb.x], 1u);
        if (bb + 1u < sThr[gb.y]) atomicAdd(&sCnt[gb.y], 1u);
        if (bb + 2u < sThr[gb.z]) atomicAdd(&sCnt[gb.z], 1u);
        if (bb + 3u < sThr[gb.w]) atomicAdd(&sCnt[gb.w], 1u);
    }
    for (; v < nvec; v += stride) {
        int4 gv = g4[v];
        uint32_t base = (uint32_t)v * 4u;
        if (base + 0u < sThr[gv.x]) atomicAdd(&sCnt[gv.x], 1u);
        if (base + 1u < sThr[gv.y]) atomicAdd(&sCnt[gv.y], 1u);
        if (base + 2u < sThr[gv.z]) atomicAdd(&sCnt[gv.z], 1u);
        if (base + 3u < sThr[gv.w]) atomicAdd(&sCnt[gv.w], 1u);
    }
    for (int i = nvec * 4 + blockIdx.x * TPB + (int)threadIdx.x; i < n; i += stride) {
        int g = gi[i];
        if ((uint32_t)i < sThr[g]) atomicAdd(&sCnt[g], 1u);
    }
    __syncthreads();

    for (int g = threadIdx.x; g < N_GROUPS; g += TPB) {
        uint32_t c = sCnt[g];
        if (c) atomicAdd((unsigned int*)&out[g], c);  // accumulate rank into d_out
    }
}

extern "C" void kernel_launch(void* const* d_in, const int* in_sizes, int n_in,
                              void* d_out, int out_size, void* d_ws, size_t ws_size,
                              hipStream_t stream) {
    const float* x  = (const float*)d_in[0];
    const int*   gi = (const int*)d_in[1];
    const int n     = in_sizes[0];
    const int nvec  = n / 4;

    unsigned long long* best = (unsigned long long*)d_ws;  // 64 KB scratch
    int* out = (int*)d_out;                                // int32 ranks, [8192]

    fa_init<<<(N_GROUPS + TPB - 1) / TPB, TPB, 0, stream>>>(best, out);
    fa_pass1<<<NB_MAIN, TPB, 0, stream>>>(x, gi, best, nvec, n);
    fa_pass2<<<NB_MAIN, TPB, 0, stream>>>(gi, best, out, nvec, n);
}